// KipfAndWillingConv_11519102288363
// MI455X (gfx1250) — compile-verified
//
#include <hip/hip_runtime.h>
#include <hip/hip_bf16.h>

typedef __attribute__((ext_vector_type(2))) float v2f;
typedef __attribute__((ext_vector_type(4))) float v4f;
typedef __attribute__((ext_vector_type(8))) float v8f;

#define NFEAT 512
#define NFILT 64
#define KCHUNK 128

// ---------------------------------------------------------------- zero
__global__ void kw_zero_kernel(float* __restrict__ p, int n) {
    int i = blockIdx.x * blockDim.x + threadIdx.x;
    if (i < n) p[i] = 0.0f;
}

// ---------------------------------------------------------------- degree histogram (native f32 atomic)
__global__ void kw_degree_kernel(const int* __restrict__ rows, float* __restrict__ deg, int E) {
    int i = blockIdx.x * blockDim.x + threadIdx.x;
    if (i < E)
        __hip_atomic_fetch_add(&deg[rows[i]], 1.0f, __ATOMIC_RELAXED, __HIP_MEMORY_SCOPE_AGENT);
}

// ---------------------------------------------------------------- deg -> dinv (in place)
__global__ void kw_dinv_kernel(float* __restrict__ deg, int N) {
    int i = blockIdx.x * blockDim.x + threadIdx.x;
    if (i < N) {
        float d = deg[i];
        deg[i] = (d > 0.0f) ? rsqrtf(d) : 0.0f;
    }
}

// ---------------------------------------------------------------- XF = X @ F via V_WMMA_F32_16X16X4_F32
// One wave computes a 16-row x 64-col stripe of XF (4 accumulators of 16x16).
// A-frag (16x4 f32): lanes 0-15 = rows M, vgpr0/1 = K+0/K+1; lanes 16-31 same rows, K+2/K+3.
// B fragments are pre-swizzled in LDS so each lane reads its {b0,b1} pair with ONE ds_load_b64:
//   sB[(g*4 + n)*64 + lane*2 + j] = F[kc + g*4 + 2*(lane>>4) + j][n*16 + (lane&15)]
// -> inner-loop B access = base + lane*8 + imm, conflict-free, no register repacking.
__global__ __launch_bounds__(256) void kw_gemm_wmma(const float* __restrict__ x,
                                                    const float* __restrict__ flt,
                                                    float* __restrict__ XF, int N) {
    __shared__ float sB[KCHUNK * NFILT];   // 32 KB, swizzled B-fragment layout

    const int lane = threadIdx.x & 31;
    const int wib  = threadIdx.x >> 5;
    const int half = lane >> 4;   // 0: K+0/1, 1: K+2/3
    const int l16  = lane & 15;

    const int stripe = blockIdx.x * 8 + wib;
    const int m0     = stripe * 16;
    const bool active = (m0 < N);

    int row = m0 + l16;
    if (row >= N) row = N - 1;
    const float* xrow = x + (size_t)row * NFEAT;

    v8f acc0 = {}, acc1 = {}, acc2 = {}, acc3 = {};

    const float* bbase = sB + lane * 2;

    for (int kc = 0; kc < NFEAT; kc += KCHUNK) {
        __syncthreads();
        // cooperative swizzled stage of filters[kc..kc+KCHUNK) x 64 into LDS
        {
            const v4f* src = (const v4f*)(flt + (size_t)kc * NFILT);
            for (int t4 = threadIdx.x; t4 < KCHUNK * (NFILT / 4); t4 += 256) {
                v4f v = src[t4];
                int r    = t4 >> 4;          // K row within chunk
                int g    = r >> 2;           // K-group of 4
                int koff = r & 3;
                int h    = koff >> 1;
                int j    = koff & 1;
                int cb   = (t4 & 15) << 2;   // col base (0..60, step 4)
                int n    = cb >> 4;          // N-tile
                int c16  = cb & 15;
                float* d = sB + (((g << 2) + n) << 6) + (((h << 4) + c16) << 1) + j;
                d[0] = v.x; d[2] = v.y; d[4] = v.z; d[6] = v.w;
            }
        }
        __syncthreads();

        if (active) {
#pragma unroll 4
            for (int k2 = 0; k2 < KCHUNK; k2 += 4) {
                // A fragment: 2 consecutive K values per lane
                v2f a = *(const v2f*)(xrow + kc + k2 + 2 * half);

                const int g = k2 >> 2;
                const float* bg = bbase + (g << 8);      // g*4*64 floats
                v2f b0 = *(const v2f*)(bg);
                v2f b1 = *(const v2f*)(bg + 64);
                v2f b2 = *(const v2f*)(bg + 128);
                v2f b3 = *(const v2f*)(bg + 192);

                acc0 = __builtin_amdgcn_wmma_f32_16x16x4_f32(false, a, false, b0, (short)0, acc0, false, false);
                acc1 = __builtin_amdgcn_wmma_f32_16x16x4_f32(false, a, false, b1, (short)0, acc1, false, false);
                acc2 = __builtin_amdgcn_wmma_f32_16x16x4_f32(false, a, false, b2, (short)0, acc2, false, false);
                acc3 = __builtin_amdgcn_wmma_f32_16x16x4_f32(false, a, false, b3, (short)0, acc3, false, false);
            }
        }
    }

    if (active) {
        // D layout: vgpr r holds row M=r (lanes 0-15) / M=r+8 (lanes 16-31), N = lane%16
#pragma unroll
        for (int r = 0; r < 8; ++r) {
            int M = m0 + r + 8 * half;
            if (M < N) {
                float* o = XF + (size_t)M * NFILT + l16;
                o[0]  = acc0[r];
                o[16] = acc1[r];
                o[32] = acc2[r];
                o[48] = acc3[r];
            }
        }
    }
}

// ---------------------------------------------------------------- edge scatter: one wave per edge
__global__ __launch_bounds__(256) void kw_scatter_kernel(const int* __restrict__ rows,
                                                         const int* __restrict__ cols,
                                                         const float* __restrict__ dinv,
                                                         const float* __restrict__ XF,
                                                         float* __restrict__ out, int E) {
    const int wid  = blockIdx.x * (blockDim.x >> 5) + (threadIdx.x >> 5);
    const int lane = threadIdx.x & 31;
    if (wid >= E) return;

    const int r = rows[wid];
    const int c = cols[wid];
    const float w = dinv[r] * dinv[c];

    const float* src = XF + (size_t)c * NFILT;
    float v0 = src[lane];
    float v1 = src[lane + 32];

    float* dst = out + (size_t)r * NFILT;
    __hip_atomic_fetch_add(&dst[lane],      w * v0, __ATOMIC_RELAXED, __HIP_MEMORY_SCOPE_AGENT);
    __hip_atomic_fetch_add(&dst[lane + 32], w * v1, __ATOMIC_RELAXED, __HIP_MEMORY_SCOPE_AGENT);
}

// ---------------------------------------------------------------- launch
extern "C" void kernel_launch(void* const* d_in, const int* in_sizes, int n_in,
                              void* d_out, int out_size, void* d_ws, size_t ws_size,
                              hipStream_t stream) {
    const float* x   = (const float*)d_in[0];
    const float* flt = (const float*)d_in[1];
    const int*   eid = (const int*)d_in[2];

    const int N = in_sizes[0] / NFEAT;
    const int E = in_sizes[2] / 2;
    const int* rows = eid;
    const int* cols = eid + E;

    float* out = (float*)d_out;
    float* deg = (float*)d_ws;                            // N floats; becomes dinv in place
    float* XF  = (float*)d_ws + (size_t)((N + 63) & ~63); // N*64 floats

    // zero accumulation buffers (d_out / d_ws are poisoned)
    kw_zero_kernel<<<(N + 255) / 256, 256, 0, stream>>>(deg, N);
    kw_zero_kernel<<<(out_size + 255) / 256, 256, 0, stream>>>(out, out_size);

    kw_degree_kernel<<<(E + 255) / 256, 256, 0, stream>>>(rows, deg, E);
    kw_dinv_kernel<<<(N + 255) / 256, 256, 0, stream>>>(deg, N);

    const int stripes = (N + 15) / 16;
    kw_gemm_wmma<<<(stripes + 7) / 8, 256, 0, stream>>>(x, flt, XF, N);

    kw_scatter_kernel<<<(E + 7) / 8, 256, 0, stream>>>(rows, cols, deg, XF, out, E);
}